// NCEAverageFull_20864951124341
// MI455X (gfx1250) — compile-verified
//
#include <hip/hip_runtime.h>
#include <math.h>
#include <stdint.h>

typedef float v2f __attribute__((ext_vector_type(2)));
typedef float v8f __attribute__((ext_vector_type(8)));

constexpr int       kB   = 256;
constexpr int       kD   = 128;
constexpr long long kN   = 262144;
constexpr int       COLS = 128;    // memory-bank rows (output columns) per block
constexpr int       CT   = COLS / 16;  // column tiles per wave
constexpr int       TPB  = 512;    // 16 wave32 waves
constexpr float     kInvT = 1.0f / 0.07f;

__global__ void nce_zero_ws(float* ws) {
    if (threadIdx.x < 2) ws[threadIdx.x] = 0.0f;
}

// One block: all 256 batch rows x 128 bank rows. Wave w owns batch rows
// [16w, 16w+16) and keeps 8 accumulators covering the 8 column sub-tiles.
// WRITE=false: accumulate sum(exp(score/T)) into ws[pair].
// WRITE=true : write exp(score/T) * (B / ws[pair]) to out.
template<bool WRITE>
__global__ __launch_bounds__(TPB)
void nce_gemm(const float* __restrict__ xl, const float* __restrict__ xab,
              const float* __restrict__ ml, const float* __restrict__ mab,
              float* __restrict__ ws, float* __restrict__ out)
{
    __shared__ float sM[COLS * kD];                 // 64 KB tile of the memory bank
    const int pair = blockIdx.y;                    // 0: out_l = l @ memory_ab.T
    const float* __restrict__ X = pair ? xab : xl;  // 1: out_ab = ab @ memory_l.T
    const float* __restrict__ M = pair ? ml  : mab;
    const long long n0 = (long long)blockIdx.x * COLS;

    // Stage M tile into LDS with CDNA5 async global->LDS copies (ASYNCcnt).
    {
        const char* gbase = (const char*)(M + n0 * kD);
        const unsigned ldsbase = (unsigned)(uintptr_t)&sM[0];
        #pragma unroll
        for (int i = 0; i < (COLS * kD * 4) / (TPB * 16); ++i) {   // 8 iters x 16B/lane
            const unsigned off = (unsigned)(i * TPB + threadIdx.x) * 16u;
            const unsigned lds = ldsbase + off;
            const char*    g   = gbase + off;
            asm volatile("global_load_async_to_lds_b128 %0, %1, off"
                         :: "v"(lds), "v"(g) : "memory");
        }
        asm volatile("s_wait_asynccnt 0x0" ::: "memory");
    }
    __syncthreads();

    const int lane = threadIdx.x & 31;
    const int wave = threadIdx.x >> 5;
    const int l16  = lane & 15;
    const int lh   = lane >> 4;        // f32 frag: lanes 0-15 hold K+{0,1}, 16-31 hold K+{2,3}
    const int rowBase = wave * 16;

    const float* __restrict__ xrow  = X  + (rowBase + l16) * kD + lh * 2;
    const float* __restrict__ bbase = sM + l16 * kD + lh * 2;

    v8f acc[CT] = {};

    #pragma unroll
    for (int k0 = 0; k0 < kD; k0 += 4) {
        const v2f a = *(const v2f*)(xrow + k0);        // A frag: 16x4 of x
        #pragma unroll
        for (int c = 0; c < CT; ++c) {
            const v2f b = *(const v2f*)(bbase + c * 16 * kD + k0);  // B frag: 4x16 of M^T
            acc[c] = __builtin_amdgcn_wmma_f32_16x16x4_f32(
                false, a, false, b, (short)0, acc[c], false, false);
        }
    }

    if (WRITE) {
        const float scale = (float)kB / ws[pair];      // 1/Z, Z = sum/B
        float* __restrict__ op = out + (long long)pair * kB * kN;
        #pragma unroll
        for (int c = 0; c < CT; ++c) {
            const long long col = n0 + c * 16 + l16;
            #pragma unroll
            for (int j = 0; j < 8; ++j) {
                // C/D frag: VGPR j holds row (rowBase + j + 8*lh), col = n0 + c*16 + lane%16
                const long long row = rowBase + j + lh * 8;
                const float e = __expf(acc[c][j] * kInvT) * scale;
                __builtin_nontemporal_store(e, &op[row * kN + col]);
            }
        }
    } else {
        float p = 0.0f;
        #pragma unroll
        for (int c = 0; c < CT; ++c)
            #pragma unroll
            for (int j = 0; j < 8; ++j)
                p += __expf(acc[c][j] * kInvT);
        // wave32 butterfly reduce
        #pragma unroll
        for (int o = 16; o > 0; o >>= 1) p += __shfl_xor(p, o, 32);
        // Block reduce reusing sM[0] (all waves are past their LDS reads).
        __syncthreads();
        if (threadIdx.x == 0) sM[0] = 0.0f;
        __syncthreads();
        if (lane == 0) atomicAdd(&sM[0], p);           // ds_add_f32
        __syncthreads();
        if (threadIdx.x == 0) atomicAdd(&ws[pair], sM[0]);
    }
}

// Swap out[b,0] <-> out[b,y[b]] for both outputs (sum-preserving, applied last).
__global__ void nce_swap(const int* __restrict__ y, float* __restrict__ out) {
    const int t = threadIdx.x;           // 512 threads: 2 pairs x 256 batch rows
    const int pair = t >> 8;
    const int b = t & 255;
    const long long off = (long long)pair * kB * kN + (long long)b * kN;
    const int yy = y[b];
    const float v0 = out[off];
    const float vy = out[off + yy];
    out[off] = vy;
    out[off + yy] = v0;
}

extern "C" void kernel_launch(void* const* d_in, const int* in_sizes, int n_in,
                              void* d_out, int out_size, void* d_ws, size_t ws_size,
                              hipStream_t stream) {
    (void)in_sizes; (void)n_in; (void)out_size; (void)ws_size;
    const float* l    = (const float*)d_in[0];
    const float* ab   = (const float*)d_in[1];
    const int*   y    = (const int*)  d_in[2];
    const float* ml   = (const float*)d_in[3];
    const float* mab  = (const float*)d_in[4];
    float* out = (float*)d_out;
    float* ws  = (float*)d_ws;

    dim3 grid((unsigned)(kN / COLS), 2);

    nce_zero_ws<<<1, 32, 0, stream>>>(ws);
    nce_gemm<false><<<grid, TPB, 0, stream>>>(l, ab, ml, mab, ws, out);
    nce_gemm<true ><<<grid, TPB, 0, stream>>>(l, ab, ml, mab, ws, out);
    nce_swap<<<1, TPB, 0, stream>>>(y, out);
}